// EdgeDistance2grid_23759759081731
// MI455X (gfx1250) — compile-verified
//
#include <hip/hip_runtime.h>
#include <hip/hip_bf16.h>

// EdgeDistance2grid: B=1, N=4096, K=32, A=4  ->  out (131072, 1344) f32 (704 MB)
// Store-bandwidth bound: 704 MB @ 23.3 TB/s ~= 30 us floor.
//
// CDNA5 paths used:
//  - v_wmma_f32_16x16x4_f32: one WMMA computes the Gram matrices of TWO pairs
//    (16 atoms x 3 coords); d^2(a,b) = G(a,a)+G(b,b)-2G(a,b)
//  - global_load_async_to_lds_b32 + s_wait_asynccnt for coord staging
//  - non-temporal b128 stores for the streamed output

#define WAVES_PER_BLOCK 8
#define RBF_INVSTD 0.95f               /* 19/20 */
#define LOG2E 1.4426950408889634f
#define LN2   0.6931471805599453f

typedef __attribute__((ext_vector_type(2))) float v2f;
typedef __attribute__((ext_vector_type(4))) float v4f;
typedef __attribute__((ext_vector_type(8))) float v8f;
typedef __attribute__((address_space(1))) int  as1_int;
typedef __attribute__((address_space(3))) int  as3_int;

#if __has_builtin(__builtin_amdgcn_wmma_f32_16x16x4_f32)
#define HAVE_WMMA 1
#else
#define HAVE_WMMA 0
#endif
#if __has_builtin(__builtin_amdgcn_global_load_async_to_lds_b32)
#define HAVE_ASYNC 1
#else
#define HAVE_ASYNC 0
#endif

__device__ __forceinline__ float rbf_term(float u, float c, float m) {
    // m * exp(-(u - c)^2)  via v_exp_f32 (2^x)
    float t = u - c;
    return m * __builtin_amdgcn_exp2f(t * t * (-LOG2E));
}

__global__ __launch_bounds__(256)
void EdgeDistance2grid_kernel(const float* __restrict__ X,        // (N, 4, 3) flat
                              const int*   __restrict__ edge_idx, // (N*K)
                              const int*   __restrict__ C,        // (N)
                              float*       __restrict__ out,      // (N*K, 1344)
                              int n_pairs) {
    // Per wave: 48 staged coords  [ i(12) | j0(12) | i(12) | j1(12) ]
    __shared__ float coords[WAVES_PER_BLOCK][48];
#if HAVE_WMMA
    __shared__ float gram[WAVES_PER_BLOCK][260];   // 16x16 Gram (+pad)
#endif

    const int tid  = (int)threadIdx.x;
    const int wave = tid >> 5;
    const int lane = tid & 31;

    // wave-global index; each wave owns pairs P0 = 2*wg and P0+1 (same node n)
    int wg = __builtin_amdgcn_readfirstlane((int)blockIdx.x * WAVES_PER_BLOCK + wave);
    const int maxwg = (n_pairs >> 1) - 1;
    const bool valid = (wg <= maxwg);
    if (wg > maxwg) wg = maxwg;                    // clamp; keeps EXEC full for WMMA

    const int P0 = wg * 2;
    const int n  = P0 >> 5;                        // k differs only in bit0 -> same n
    const int j0 = edge_idx[P0];                   // wave-uniform -> s_load
    const int j1 = edge_idx[P0 + 1];
    const int cn = C[n];
    const float mask0 = (cn > 0 && C[j0] > 0) ? 1.0f : 0.0f;
    const float mask1 = (cn > 0 && C[j1] > 0) ? 1.0f : 0.0f;

    // ---- stage 48 floats (2 pairs x 8 atoms x 3 coords) into LDS -------------
    {
        auto stage = [&](int f) {                  // f in [0,48)
            const int c    = f / 12;               // chunk: 0=i,1=j0,2=i,3=j1
            const int o    = f - c * 12;
            const int node = (c & 1) ? ((c & 2) ? j1 : j0) : n;
            const float* src = X + (size_t)node * 12 + o;
#if HAVE_ASYNC
            __builtin_amdgcn_global_load_async_to_lds_b32(
                (as1_int*)(void*)const_cast<float*>(src),
                (as3_int*)(void*)&coords[wave][f], 0, 0);
#else
            coords[wave][f] = *src;
#endif
        };
        stage(lane);
        if (lane < 16) stage(32 + lane);
    }
#if __has_builtin(__builtin_amdgcn_s_wait_asynccnt)
    __builtin_amdgcn_s_wait_asynccnt(0);
#endif
    __syncthreads();

    const float* cw = coords[wave];

    // atom m (0..15): pair g = m>>3, local atom a = m&7, coords at g*24 + 3a
    const int am  = lane & 15;
    const int aoff = (am >> 3) * 24 + (am & 7) * 3;

#if HAVE_WMMA
    // A (16x4 f32, 2 VGPRs): lanes 0-15: {x_m, y_m} (K=0,1); lanes 16-31: {z_m, 0} (K=2,3)
    // B (4x16 f32) has the identical per-lane layout for B = A^T -> reuse the same regs.
    v2f ab;
    if (lane < 16) { ab.x = cw[aoff + 0]; ab.y = cw[aoff + 1]; }
    else           { ab.x = cw[aoff + 2]; ab.y = 0.0f;         }
    v8f acc = {};
    acc = __builtin_amdgcn_wmma_f32_16x16x4_f32(
        /*neg_a=*/false, ab, /*neg_b=*/false, ab,
        /*c_mod=*/(short)0, acc, /*reuse_a=*/false, /*reuse_b=*/false);

    // spill Gram to LDS: VGPR r, lanes 0-15 -> (M=r, N=lane); lanes 16-31 -> (M=8+r, N=lane-16)
#pragma unroll
    for (int r = 0; r < 8; ++r) gram[wave][r * 32 + lane] = acc[r];
    __syncthreads();
    // entry (m, n), n absolute in [0,16): index = (m&7)*32 + (m>>3)*16 + n
#define GRAM(m, nn) gram[wave][((m) & 7) * 32 + (((m) >> 3) * 16) + (nn)]
#endif

    // ---- features: half-wave g covers its pair's 64 p values, 4 per lane -----
    // lane -> g = lane>>4, idx = lane&15 ; p = idx + 16t ; a = p>>3 ; b = idx&7 (t-invariant)
    const int   g     = lane >> 4;
    const int   idx   = lane & 15;
    const int   bb    = idx & 7;
    const float maskl = g ? mask1 : mask0;
    const size_t pbase = (size_t)(P0 + g) * 1344;

#if HAVE_WMMA
    const int   mbv    = 8 * g + bb;
    const float diag_b = GRAM(mbv, mbv);
#else
    const float bx = cw[g * 24 + bb * 3 + 0];
    const float by = cw[g * 24 + bb * 3 + 1];
    const float bz = cw[g * 24 + bb * 3 + 2];
#endif

#pragma unroll
    for (int t = 0; t < 4; ++t) {
        const int p = idx + 16 * t;
        const int a = p >> 3;
        float d2;
#if HAVE_WMMA
        const int m = 8 * g + a;
        d2 = GRAM(m, m) + diag_b - 2.0f * GRAM(m, mbv);
        d2 = fmaxf(d2, 0.0f);                     // Gram-form cancellation guard
#else
        const float dx = cw[g * 24 + a * 3 + 0] - bx;
        const float dy = cw[g * 24 + a * 3 + 1] - by;
        const float dz = cw[g * 24 + a * 3 + 2] - bz;
        d2 = dx * dx + dy * dy + dz * dz;
#endif
        const float d = __builtin_amdgcn_sqrtf(d2 + 1e-6f);
        const float u = d * RBF_INVSTD;           // (d - c*std)/std = d/std - c

        v4f* __restrict__ o = (v4f*)(out + pbase + (size_t)p * 20);  // 80p bytes: 16B aligned
#pragma unroll
        for (int q = 0; q < 5; ++q) {
            const float c = (float)(4 * q);
            v4f r;
            r.x = rbf_term(u, c + 0.0f, maskl);
            r.y = rbf_term(u, c + 1.0f, maskl);
            r.z = rbf_term(u, c + 2.0f, maskl);
            r.w = rbf_term(u, c + 3.0f, maskl);
            if (valid) __builtin_nontemporal_store(r, o + q);        // NT b128
        }
        const float lg = maskl * (LN2 * __builtin_amdgcn_logf(d + 0.01f));
        if (valid) __builtin_nontemporal_store(lg, out + pbase + 1280 + p);
    }
}

extern "C" void kernel_launch(void* const* d_in, const int* in_sizes, int n_in,
                              void* d_out, int out_size, void* d_ws, size_t ws_size,
                              hipStream_t stream) {
    const float* X        = (const float*)d_in[0];  // (B,N,A,3) f32
    const int*   edge_idx = (const int*)d_in[1];    // (B,N,K)
    const int*   C        = (const int*)d_in[2];    // (B,N)

    float* out = (float*)d_out;

    const int n_pairs = in_sizes[1];                              // B*N*K = 131072
    const int n_waves = n_pairs / 2;                              // 2 pairs per wave
    const int blocks  = (n_waves + WAVES_PER_BLOCK - 1) / WAVES_PER_BLOCK;

    EdgeDistance2grid_kernel<<<dim3(blocks), dim3(256), 0, stream>>>(
        X, edge_idx, C, out, n_pairs);
}